// CTCLayer_2095944040773
// MI455X (gfx1250) — compile-verified
//
#include <hip/hip_runtime.h>
#include <math.h>

#define NEG_INF_F (-1.0e30f)
#define EPS_F 1e-7f

constexpr int Bsz = 256;
constexpr int Tsz = 1024;
constexpr int Csz = 128;
constexpr int Lsz = 128;
constexpr int SMAX = 2 * Lsz + 1;  // 257
constexpr int BLANK = Csz - 1;     // 127

// One wave (32 lanes x 16B) asynchronously copies a whole 512B probability row
// from global memory into LDS (CDNA5 GLOBAL_LOAD_ASYNC_TO_LDS, tracked by ASYNCcnt).
__device__ __forceinline__ void async_row_load(const float* gsrc, float* lds_dst, int lane) {
  // Low 32 bits of the generic (flat) shared-pointer value are the wave-relative
  // LDS byte offset, which is what the async-LDS instruction's VDST expects.
  unsigned lds_off = (unsigned)(size_t)lds_dst + (unsigned)(lane * 16);
  unsigned long long gaddr = (unsigned long long)(size_t)(gsrc + lane * 4);
  asm volatile("global_load_async_to_lds_b128 %0, %1, off"
               :
               : "v"(lds_off), "v"(gaddr)
               : "memory");
}

__device__ __forceinline__ void wait_async() {
  asm volatile("s_wait_asynccnt 0x0" ::: "memory");
}

__global__ __launch_bounds__(256) void ctc_batch_cost_kernel(
    const int* __restrict__ y_true,      // [B, L]
    const float* __restrict__ y_pred,    // [B, T, C] (softmax probs)
    const int* __restrict__ input_length,// [B, 1]
    const int* __restrict__ label_length,// [B, 1]
    float* __restrict__ out)             // [B, 1]
{
  const int b = blockIdx.x;
  const int tid = threadIdx.x;

  __shared__ float rowbuf[2][Csz];       // double-buffered prob rows
  __shared__ float alpha[2][SMAX + 2];   // +2 leading NEG_INF padding slots

  const int il = input_length[b];
  const int ll = label_length[b];
  const int smax_valid = 2 * ll + 1;

  const float* yp = y_pred + (size_t)b * Tsz * Csz;
  const int* yt = y_true + (size_t)b * Lsz;

  // Per-state metadata; thread tid owns state s0 = tid, thread 0 also owns s=256.
  const int s0 = tid;
  int ext0;
  bool skip0;
  if (s0 & 1) {
    const int lab = yt[(s0 - 1) >> 1];
    ext0 = lab;
    const int prev = (s0 >= 3) ? yt[(s0 - 3) >> 1] : -1;
    skip0 = (s0 >= 3) && (lab != BLANK) && (lab != prev);
  } else {
    ext0 = BLANK;
    skip0 = false;
  }

  // alpha at t=0
  if (tid < 2) {
    alpha[0][tid] = NEG_INF_F;
    alpha[1][tid] = NEG_INF_F;
  }
  {
    float a0 = NEG_INF_F;
    if (s0 == 0) a0 = logf(yp[BLANK] + EPS_F);
    else if (s0 == 1 && ll > 0) a0 = logf(yp[yt[0]] + EPS_F);
    alpha[0][s0 + 2] = a0;
    if (tid == 0) alpha[0][2 * Lsz + 2] = NEG_INF_F;  // s = 256
  }

  // Preload probability row t=1 into rowbuf[1].
  if (tid < 32) {
    if (il > 1) async_row_load(yp + Csz, &rowbuf[1][0], tid);
    wait_async();
  }
  __syncthreads();

  int cu = 0;  // alpha[cu] = alpha at time t-1
  for (int t = 1; t < il; ++t) {
    const int tb = t & 1;
    const int nx = cu ^ 1;

    // Kick off async fetch of row t+1 while computing step t; prefetch t+8 into L2.
    if (tid < 32) {
      const int tn = t + 1;
      if (tn < il) async_row_load(yp + (size_t)tn * Csz, &rowbuf[tb ^ 1][0], tid);
      const int tp = t + 8;
      if (tp < il && tid < 4)
        __builtin_prefetch(yp + (size_t)tp * Csz + tid * 32, 0, 1);
    }

    // Owned state s0.
    {
      const float a1 = alpha[cu][s0 + 2];
      const float a2 = alpha[cu][s0 + 1];
      const float a3 = skip0 ? alpha[cu][s0] : NEG_INF_F;
      const float m = fmaxf(a1, fmaxf(a2, a3));
      const float lp = logf(rowbuf[tb][ext0] + EPS_F);
      float nv = m + logf(expf(a1 - m) + expf(a2 - m) + expf(a3 - m)) + lp;
      if (s0 >= smax_valid) nv = NEG_INF_F;
      alpha[nx][s0 + 2] = nv;
    }
    // State s = 256 (even -> blank, no skip), handled by thread 0.
    if (tid == 0) {
      const int s = 2 * Lsz;
      const float a1 = alpha[cu][s + 2];
      const float a2 = alpha[cu][s + 1];
      const float a3 = NEG_INF_F;
      const float m = fmaxf(a1, fmaxf(a2, a3));
      const float lp = logf(rowbuf[tb][BLANK] + EPS_F);
      float nv = m + logf(expf(a1 - m) + expf(a2 - m) + expf(a3 - m)) + lp;
      if (s >= smax_valid) nv = NEG_INF_F;
      alpha[nx][s + 2] = nv;
    }

    if (tid < 32) wait_async();  // row t+1 resident in LDS
    __syncthreads();             // publishes alpha[nx] + rowbuf to all waves
    cu = nx;
  }

  if (tid == 0) {
    const int send = 2 * ll;
    const float a_end = alpha[cu][send + 2];
    const float a_pen = (ll > 0) ? alpha[cu][send + 1] : NEG_INF_F;
    const float m = fmaxf(a_end, a_pen);
    const float lo = fminf(a_end, a_pen);
    out[b] = -(m + log1pf(expf(lo - m)));
  }
}

extern "C" void kernel_launch(void* const* d_in, const int* in_sizes, int n_in,
                              void* d_out, int out_size, void* d_ws, size_t ws_size,
                              hipStream_t stream) {
  (void)in_sizes; (void)n_in; (void)out_size; (void)d_ws; (void)ws_size;
  const int* y_true = (const int*)d_in[0];
  const float* y_pred = (const float*)d_in[1];
  const int* input_length = (const int*)d_in[2];
  const int* label_length = (const int*)d_in[3];
  float* out = (float*)d_out;

  ctc_batch_cost_kernel<<<dim3(Bsz), dim3(256), 0, stream>>>(
      y_true, y_pred, input_length, label_length, out);
}